// AFM_84456236908948
// MI455X (gfx1250) — compile-verified
//
#include <hip/hip_runtime.h>

// AFM reference collapsed (exact for finite inputs):
//  - softmax over a size-1 axis == 1.0  => Wa1/ba1/Wa2/ba2 are dead inputs
//  - sum_{i<j} (d_i v_i)(d_j v_j) == ((sum_n y)^2 - sum_n y^2)/2, y = d*v
// out[b] = sum_n dense[b,n,0]*Wl[n] + bl + sum_m (Wp[m]/2P)*(s_m^2 - q_m) + bp
//
// HBM-bound: one 32 MiB pass over dense at 23.3 TB/s (~1.4 us floor).
//  - lane p loads dense[n, 4p..4p+3] as B128; element e belongs to m-chunk e
//    (m = 4p + e), column p of the WMMA tile.
//  - VALU (v_pk_fma_f32) pre-accumulates over 8 n-rows per (half, slot)
//    K-position, then ONE v_wmma_f32_16x16x4_f32 with all-ones A finishes the
//    32-lane column sum (K0: n0-7, K1: n8-15, K2: n16-23, K3: n24-31).
//  - Duplicated D rows across lane halves folded into the 1/(2P) scale (x0.5).
//  - Final 32-lane reduce via raw ds_swizzle_b32 XOR steps (no bounds checks).
//  - v/Wl/Wp/biases hoisted; 4 batches per wave; prefetch next batch.

typedef __attribute__((ext_vector_type(2))) float v2f;
typedef __attribute__((ext_vector_type(4))) float v4f;
typedef __attribute__((ext_vector_type(8))) float v8f;

#define AFM_N 32
#define AFM_M 64
#define AFM_NPAIR 496         // 32*31/2
#define BATCHES_PER_WAVE 4

// ds_swizzle group-of-32 XOR-mode: offset = (xor_mask << 10) | 0x1f
#define SWZ_ADD(x, OFF)                                                        \
    ((x) + __int_as_float(__builtin_amdgcn_ds_swizzle(__float_as_int(x), OFF)))

__global__ __launch_bounds__(256) void AFM_fused_kernel(
    const float* __restrict__ dense,  // (B, 32, 64)
    const float* __restrict__ Wl,     // (32, 1)
    const float* __restrict__ bl,     // (1,)
    const float* __restrict__ v,      // (32, 64)
    const float* __restrict__ Wp,     // (64, 1)
    const float* __restrict__ bp,     // (1,)
    float* __restrict__ out,          // (B, 1)
    int B)
{
    const int lane   = threadIdx.x & 31;
    const int waveId = blockIdx.x * (blockDim.x >> 5) + (threadIdx.x >> 5);

    const int half = lane >> 4;    // K2/K3 vs K0/K1 striping half
    const int p    = lane & 15;    // WMMA column index

    // K-position n-groups for this lane's two B-operand VGPR slots:
    //   slot 0 -> K = 2*half     -> n in [ (2*half+0)*8, +8 )
    //   slot 1 -> K = 2*half + 1 -> n in [ (2*half+1)*8, +8 )
    const int nBase0 = (half * 2 + 0) * 8;
    const int nBase1 = (half * 2 + 1) * 8;

    // ---- hoisted loop invariants (shared weights, L2-resident) ----
    v4f vReg[2][8];
    #pragma unroll
    for (int r = 0; r < 8; ++r) {
        vReg[0][r] = *(const v4f*)(v + (nBase0 + r) * AFM_M + 4 * p);
        vReg[1][r] = *(const v4f*)(v + (nBase1 + r) * AFM_M + 4 * p);
    }
    const v4f  wpReg = *(const v4f*)(Wp + 4 * p);   // Wp[m] for m = 4p+e
    const float wl   = Wl[lane];                    // lane == n for linear term
    const float bias = bl[0] + bp[0];
    // 1/(2P), times 0.5 because prod is duplicated across both lane halves
    const float cS   = 0.25f / (float)AFM_NPAIR;

    v2f onesA; onesA[0] = 1.0f; onesA[1] = 1.0f;    // all-ones A matrix

    const int bStart = waveId * BATCHES_PER_WAVE;

    for (int bi = 0; bi < BATCHES_PER_WAVE; ++bi) {
        const int b = bStart + bi;
        if (b >= B) break;                          // wave-uniform: EXEC all-1s
        const float* __restrict__ db = dense + (size_t)b * (AFM_N * AFM_M);

        // Prefetch next batch's stream into L2 (global_prefetch_b8)
        if (bi + 1 < BATCHES_PER_WAVE && b + 1 < B) {
            const float* nb = db + AFM_N * AFM_M;
            __builtin_prefetch(nb + nBase0 * AFM_M + 4 * p, 0, 1);
            __builtin_prefetch(nb + nBase1 * AFM_M + 4 * p, 0, 1);
        }

        // ---- VALU pre-accumulation over this lane's 16 n-rows ----
        v4f accS0 = {0.f,0.f,0.f,0.f}, accS1 = {0.f,0.f,0.f,0.f};
        v4f accQ0 = {0.f,0.f,0.f,0.f}, accQ1 = {0.f,0.f,0.f,0.f};
        #pragma unroll
        for (int r = 0; r < 8; ++r) {
            const v4f d0 = *(const v4f*)(db + (nBase0 + r) * AFM_M + 4 * p);
            const v4f d1 = *(const v4f*)(db + (nBase1 + r) * AFM_M + 4 * p);
            const v4f y0 = d0 * vReg[0][r];
            const v4f y1 = d1 * vReg[1][r];
            accS0 += y0;          accS1 += y1;
            accQ0 += y0 * y0;     accQ1 += y1 * y1;   // v_pk_fma_f32
        }

        // ---- WMMA cross-lane column sums, one (s,q) pair per m-chunk ----
        float prod = 0.0f;
        #pragma unroll
        for (int e = 0; e < 4; ++e) {                // chunk e: m = 4p + e
            v2f Bs; Bs[0] = accS0[e]; Bs[1] = accS1[e];
            v2f Bq; Bq[0] = accQ0[e]; Bq[1] = accQ1[e];
            v8f zero8 = {0.f,0.f,0.f,0.f,0.f,0.f,0.f,0.f};
            v8f sD = __builtin_amdgcn_wmma_f32_16x16x4_f32(
                         false, onesA, false, Bs, (short)0, zero8, false, false);
            v8f qD = __builtin_amdgcn_wmma_f32_16x16x4_f32(
                         false, onesA, false, Bq, (short)0, zero8, false, false);
            const float s = sD[0];                   // every D row == column sums
            const float q = qD[0];
            prod += wpReg[e] * (s * s - q);          // duplicated on both halves
        }

        // ---- linear term (lane == n); line already in L0 from B128 loads ----
        float red = prod * cS + db[lane * AFM_M + 0] * wl;

        // ---- full 32-lane reduce: 5 x (ds_swizzle XOR + add) ----
        red = SWZ_ADD(red, 0x401f);   // xor 16
        red = SWZ_ADD(red, 0x201f);   // xor 8
        red = SWZ_ADD(red, 0x101f);   // xor 4
        red = SWZ_ADD(red, 0x081f);   // xor 2
        red = SWZ_ADD(red, 0x041f);   // xor 1

        if (lane == 0)
            out[b] = red + bias;
    }
}

extern "C" void kernel_launch(void* const* d_in, const int* in_sizes, int n_in,
                              void* d_out, int out_size, void* d_ws, size_t ws_size,
                              hipStream_t stream) {
    // setup_inputs order: 0:dense 1:Wl 2:bl 3:v 4:Wa1 5:ba1 6:Wa2 7:ba2 8:Wp 9:bp
    const float* dense = (const float*)d_in[0];
    const float* Wl    = (const float*)d_in[1];
    const float* bl    = (const float*)d_in[2];
    const float* v     = (const float*)d_in[3];
    const float* Wp    = (const float*)d_in[8];
    const float* bp    = (const float*)d_in[9];
    float* out = (float*)d_out;

    const int B = in_sizes[0] / (AFM_N * AFM_M);    // 4096

    const int threads       = 256;                  // 8 wave32 per block
    const int wavesPerBlock = threads / 32;
    const int totalWaves    = (B + BATCHES_PER_WAVE - 1) / BATCHES_PER_WAVE;
    const int blocks        = (totalWaves + wavesPerBlock - 1) / wavesPerBlock;

    AFM_fused_kernel<<<blocks, threads, 0, stream>>>(dense, Wl, bl, v, Wp, bp, out, B);
}